// SSMLayer_14877766713395
// MI455X (gfx1250) — compile-verified
//
#include <hip/hip_runtime.h>
#include <hip/hip_bf16.h>
#include <math.h>

// SSM layer for MI455X (gfx1250, wave32, WMMA, async-to-LDS DMA).
// Pipeline: setup -> Bu GEMM (WMMA f32 16x16x4, double-buffered async LDS staging)
//        -> chunked scan (3 passes)
//        -> output GEMM (WMMA f32 16x16x4, async LDS staging) fused with D*u, NT stores.

#define BATCH   8
#define SEQ     4096
#define I_DIM   1024
#define N_DIM   64
#define R_DIM   (BATCH * SEQ)   // 32768 flattened rows
#define CHUNK   64              // scan chunk length
#define NCHUNK  (SEQ / CHUNK)   // 64 chunks
#define KTILES  (I_DIM / 32)    // 32 K-tiles in Bu GEMM

typedef __attribute__((ext_vector_type(2))) float v2f;
typedef __attribute__((ext_vector_type(8))) float v8f;

// Low 32 bits of a flat pointer into the LDS aperture == workgroup-relative LDS
// byte address -> usable as async-op VDST.
__device__ __forceinline__ unsigned lds_addr32(const void* p) {
    return (unsigned)(uintptr_t)p;
}

__device__ __forceinline__ void async_load_b128(unsigned lds_off, const void* gaddr) {
    asm volatile("global_load_async_to_lds_b128 %0, %1, off"
                 :: "v"(lds_off), "v"(gaddr) : "memory");
}

// ---------------------------------------------------------------------------
// Kernel 0: delta = softplus(log_delta); abar[n] = mean_i exp(delta_i * -exp(log_A_n))
// ---------------------------------------------------------------------------
__global__ __launch_bounds__(1024) void ssm_setup(const float* __restrict__ log_A,
                                                  const float* __restrict__ log_delta,
                                                  float* __restrict__ delta,
                                                  float* __restrict__ abar) {
    __shared__ float sd[I_DIM];
    const int t = threadIdx.x;
    float x = log_delta[t];
    float d = (x > 20.f) ? x : log1pf(expf(x));   // softplus
    sd[t] = d;
    delta[t] = d;
    __syncthreads();
    if (t < N_DIM) {
        float A = -expf(log_A[t]);
        float s = 0.f;
        for (int i = 0; i < I_DIM; ++i) s += expf(sd[i] * A);
        abar[t] = s * (1.0f / (float)I_DIM);
    }
}

// ---------------------------------------------------------------------------
// Kernel 1: Bu[r, n] = sum_i (B[n,i] * delta[i]) * u[r, i]
// Block = 256 threads (8 waves). Block tile: 128 rows x 64 states.
// u tile staged by GLOBAL_LOAD_ASYNC_TO_LDS_B128, double-buffered so the
// next tile's DMA overlaps the current tile's WMMAs.
// Stride 36 floats: keeps every b128 LDS destination 16B-aligned and gives
// distinct banks across the 16 fragment rows.
// ---------------------------------------------------------------------------
__global__ __launch_bounds__(256) void ssm_bu_gemm(const float* __restrict__ u,
                                                   const float* __restrict__ Bmat,
                                                   const float* __restrict__ delta,
                                                   float* __restrict__ bu) {
    __shared__ float lds_u[2][128 * 36];   // [buf][row][k]
    __shared__ float lds_b[2][64 * 36];    // [buf][n][k]  (B_bar, k-fast)

    const int tid  = threadIdx.x;
    const int wave = tid >> 5;
    const int lane = tid & 31;
    const int lm   = lane & 15;      // M / N index within 16
    const int lh   = lane >> 4;      // K half-select
    const int r0   = blockIdx.x * 128;

    // 4 async b128 per thread cover the 128x32 u tile.
    const int ld_row = tid >> 3;         // 0..31 (base row within 32-row groups)
    const int ld_kk  = (tid & 7) * 4;    // 0..28

    auto issue_u_tile = [&](int k0, int buf) {
        #pragma unroll
        for (int g = 0; g < 4; ++g) {
            const int row = g * 32 + ld_row;
            async_load_b128(lds_addr32(&lds_u[buf][row * 36 + ld_kk]),
                            u + (size_t)(r0 + row) * I_DIM + k0 + ld_kk);
        }
    };

    v8f acc[4] = {};

    issue_u_tile(0, 0);

    for (int it = 0; it < KTILES; ++it) {
        const int k0  = it * 32;
        const int buf = it & 1;

        if (it + 1 < KTILES) issue_u_tile(k0 + 32, buf ^ 1);

        // --- stage B_bar tile: 64 n x 32 k, scaled by delta on the fly ---
        {
            const int n  = tid >> 2;        // 0..63
            const int kk = (tid & 3) * 8;   // 0,8,16,24
            const float4 b0 = *(const float4*)(Bmat + (size_t)n * I_DIM + k0 + kk);
            const float4 b1 = *(const float4*)(Bmat + (size_t)n * I_DIM + k0 + kk + 4);
            const float4 d0 = *(const float4*)(delta + k0 + kk);
            const float4 d1 = *(const float4*)(delta + k0 + kk + 4);
            float* dst = &lds_b[buf][n * 36 + kk];
            v2f t0; t0.x = b0.x * d0.x; t0.y = b0.y * d0.y;
            v2f t1; t1.x = b0.z * d0.z; t1.y = b0.w * d0.w;
            v2f t2; t2.x = b1.x * d1.x; t2.y = b1.y * d1.y;
            v2f t3; t3.x = b1.z * d1.z; t3.y = b1.w * d1.w;
            *(v2f*)(dst + 0) = t0;
            *(v2f*)(dst + 2) = t1;
            *(v2f*)(dst + 4) = t2;
            *(v2f*)(dst + 6) = t3;
        }

        // Async loads complete in order: waiting <=4 guarantees the current
        // tile's 4 transfers are done while next tile's 4 stay in flight.
        if (it + 1 < KTILES)
            asm volatile("s_wait_asynccnt 0x4" ::: "memory");
        else
            asm volatile("s_wait_asynccnt 0x0" ::: "memory");
        __syncthreads();

        // --- WMMA: 8 k-steps x 4 n-tiles ---
        #pragma unroll
        for (int kk = 0; kk < 32; kk += 4) {
            const int off = kk + 2 * lh;   // lane K position within 4-wide step
            const v2f a = *(const v2f*)&lds_u[buf][(wave * 16 + lm) * 36 + off];
            #pragma unroll
            for (int nt = 0; nt < 4; ++nt) {
                const v2f b = *(const v2f*)&lds_b[buf][(nt * 16 + lm) * 36 + off];
                acc[nt] = __builtin_amdgcn_wmma_f32_16x16x4_f32(
                    false, a, false, b, (short)0, acc[nt], false, false);
            }
        }
        __syncthreads();
    }

    // Store Bu (C/D layout: vgpr v -> row v + 8*lh, col = lm within tile)
    #pragma unroll
    for (int nt = 0; nt < 4; ++nt) {
        #pragma unroll
        for (int v = 0; v < 8; ++v) {
            const int row = r0 + wave * 16 + v + 8 * lh;
            const int col = nt * 16 + lm;
            bu[(size_t)row * N_DIM + col] = acc[nt][v];
        }
    }
}

// ---------------------------------------------------------------------------
// Kernel 2a: chunk-local scan (x starts at 0 per chunk), in place on bu.
// ---------------------------------------------------------------------------
__global__ __launch_bounds__(64) void ssm_scan_local(float* __restrict__ bu,
                                                     const float* __restrict__ abar) {
    const int b = blockIdx.x >> 6;
    const int c = blockIdx.x & (NCHUNK - 1);
    const int n = threadIdx.x;
    const float a = abar[n];
    float x = 0.f;
    size_t base = ((size_t)b * SEQ + (size_t)c * CHUNK) * N_DIM + n;
    for (int t = 0; t < CHUNK; ++t) {
        x = fmaf(x, a, bu[base]);
        bu[base] = x;
        base += N_DIM;
    }
}

// ---------------------------------------------------------------------------
// Kernel 2b: scan across chunk-final states -> carry-in per chunk.
// ---------------------------------------------------------------------------
__global__ __launch_bounds__(256) void ssm_scan_carry(const float* __restrict__ bu,
                                                      const float* __restrict__ abar,
                                                      float* __restrict__ carry) {
    const int idx = blockIdx.x * blockDim.x + threadIdx.x;  // 0..511
    const int b = idx >> 6;
    const int n = idx & (N_DIM - 1);
    const float a = abar[n];
    float aL = a;
    #pragma unroll
    for (int i = 0; i < 6; ++i) aL *= aL;   // a^64
    float s = 0.f;
    for (int c = 0; c < NCHUNK; ++c) {
        carry[((size_t)b * NCHUNK + c) * N_DIM + n] = s;
        const float fin = bu[((size_t)b * SEQ + (size_t)c * CHUNK + (CHUNK - 1)) * N_DIM + n];
        s = fmaf(s, aL, fin);
    }
}

// ---------------------------------------------------------------------------
// Kernel 2c: fix-up: x_t(global) = x_t(local) + carry * a^(t_local+1)
// ---------------------------------------------------------------------------
__global__ __launch_bounds__(64) void ssm_scan_fix(float* __restrict__ bu,
                                                   const float* __restrict__ abar,
                                                   const float* __restrict__ carry) {
    const int b = blockIdx.x >> 6;
    const int c = blockIdx.x & (NCHUNK - 1);
    const int n = threadIdx.x;
    if (c == 0) return;
    const float a = abar[n];
    const float s = carry[((size_t)b * NCHUNK + c) * N_DIM + n];
    float f = a;
    size_t base = ((size_t)b * SEQ + (size_t)c * CHUNK) * N_DIM + n;
    for (int t = 0; t < CHUNK; ++t) {
        bu[base] = fmaf(s, f, bu[base]);
        f *= a;
        base += N_DIM;
    }
}

// ---------------------------------------------------------------------------
// Kernel 3: y[r, i] = sum_n C[i, n] * xs[r, n] + D[i] * u[r, i]
// Block tile: 128 rows x 64 i-cols, K = 64 states in one shot.
// Tiles staged via async-to-LDS DMA; u re-read should hit the 192MB L2;
// y stored non-temporally so the output stream doesn't evict u.
// Stride 68 floats keeps b128 LDS destinations 16B-aligned.
// ---------------------------------------------------------------------------
__global__ __launch_bounds__(256) void ssm_y_gemm(const float* __restrict__ xs,
                                                  const float* __restrict__ Cmat,
                                                  const float* __restrict__ Dvec,
                                                  const float* __restrict__ u,
                                                  float* __restrict__ y) {
    __shared__ float lds_x[128 * 68];   // [row][k]
    __shared__ float lds_c[64 * 68];    // [i][k]   (C is already i-major, k-fast)

    const int tid  = threadIdx.x;
    const int wave = tid >> 5;
    const int lane = tid & 31;
    const int lm   = lane & 15;
    const int lh   = lane >> 4;
    const int r0   = blockIdx.x * 128;
    const int i0   = blockIdx.y * 64;

    // stage xs tile: 128 rows x 64 states (8 async b128 per thread)
    #pragma unroll
    for (int j = 0; j < 8; ++j) {
        const int e   = tid + j * 256;
        const int row = e >> 4;
        const int kk  = (e & 15) * 4;
        async_load_b128(lds_addr32(&lds_x[row * 68 + kk]),
                        xs + (size_t)(r0 + row) * N_DIM + kk);
    }
    // stage C tile: 64 i x 64 states (4 async b128 per thread)
    #pragma unroll
    for (int j = 0; j < 4; ++j) {
        const int e  = tid + j * 256;
        const int ii = e >> 4;
        const int kk = (e & 15) * 4;
        async_load_b128(lds_addr32(&lds_c[ii * 68 + kk]),
                        Cmat + (size_t)(i0 + ii) * N_DIM + kk);
    }
    asm volatile("s_wait_asynccnt 0x0" ::: "memory");
    __syncthreads();

    v8f acc[4] = {};
    #pragma unroll
    for (int kk = 0; kk < N_DIM; kk += 4) {
        const int off = kk + 2 * lh;
        const v2f a = *(const v2f*)&lds_x[(wave * 16 + lm) * 68 + off];
        #pragma unroll
        for (int nt = 0; nt < 4; ++nt) {
            const v2f b = *(const v2f*)&lds_c[(nt * 16 + lm) * 68 + off];
            acc[nt] = __builtin_amdgcn_wmma_f32_16x16x4_f32(
                false, a, false, b, (short)0, acc[nt], false, false);
        }
    }

    // Epilogue: + D[i] * u[r, i]; non-temporal stores for y.
    #pragma unroll
    for (int nt = 0; nt < 4; ++nt) {
        const int i = i0 + nt * 16 + lm;
        const float d = Dvec[i];
        #pragma unroll
        for (int v = 0; v < 8; ++v) {
            const int row = r0 + wave * 16 + v + 8 * lh;
            const size_t idx = (size_t)row * I_DIM + i;
            const float val = acc[nt][v] + d * u[idx];
            __builtin_nontemporal_store(val, &y[idx]);
        }
    }
}

// ---------------------------------------------------------------------------
// Host launcher
// ---------------------------------------------------------------------------
extern "C" void kernel_launch(void* const* d_in, const int* in_sizes, int n_in,
                              void* d_out, int out_size, void* d_ws, size_t ws_size,
                              hipStream_t stream) {
    const float* u         = (const float*)d_in[0];   // [8, 4096, 1024]
    const float* log_A     = (const float*)d_in[1];   // [64]
    const float* Bmat      = (const float*)d_in[2];   // [64, 1024]
    const float* Cmat      = (const float*)d_in[3];   // [1024, 64]
    const float* Dvec      = (const float*)d_in[4];   // [1024]
    const float* log_delta = (const float*)d_in[5];   // [1024]
    float* y = (float*)d_out;

    float* w      = (float*)d_ws;
    float* delta  = w;                    // 1024
    float* abar   = w + 1024;             // 64
    float* carry  = w + 2048;             // 8*64*64 = 32768
    float* bu     = w + 36864;            // 32768*64 = 2M floats (8 MB), 16B aligned

    ssm_setup<<<1, 1024, 0, stream>>>(log_A, log_delta, delta, abar);

    ssm_bu_gemm<<<R_DIM / 128, 256, 0, stream>>>(u, Bmat, delta, bu);

    ssm_scan_local<<<BATCH * NCHUNK, 64, 0, stream>>>(bu, abar);
    ssm_scan_carry<<<2, 256, 0, stream>>>(bu, abar, carry);
    ssm_scan_fix<<<BATCH * NCHUNK, 64, 0, stream>>>(bu, abar, carry);

    dim3 grid(R_DIM / 128, I_DIM / 64);
    ssm_y_gemm<<<grid, 256, 0, stream>>>(bu, Cmat, Dvec, u, y);
}